// EdgeDecoder_7172595384374
// MI455X (gfx1250) — compile-verified
//
#include <hip/hip_runtime.h>
#include <hip/hip_bf16.h>

// CDNA5 / gfx1250, wave32. FP32 WMMA path (V_WMMA_F32_16X16X4_F32) keeps the
// reference's fp32 numerics while running the dominant [E,256]x[256,128] GEMM
// on the matrix pipe. M-tile = 64 edges/block -> each B fragment feeds 4 WMMAs,
// halving W1 L2 traffic vs the 32-edge version.

typedef __attribute__((ext_vector_type(2))) float v2f;
typedef __attribute__((ext_vector_type(8))) float v8f;

#define HID     128          // H
#define TWOH    256          // 2H (concat width, GEMM K)
#define ROWPAD  260          // LDS row pitch in floats (bank-conflict padding; 260*4 % 16 == 0)
#define M_TILE  64           // edges per block (four 16-row WMMA subtiles)

__global__ __launch_bounds__(256)
void EdgeDecoder_wmma_f32(const float* __restrict__ zu,    // [NU,128]
                          const float* __restrict__ zb,    // [NB,128]
                          const int*   __restrict__ eidx,  // [2,E]
                          const float* __restrict__ W1,    // [256,128] row-major
                          const float* __restrict__ b1,    // [128]
                          const float* __restrict__ W2,    // [128,1]
                          const float* __restrict__ b2,    // [1]
                          float*       __restrict__ out,   // [E]
                          int E)
{
    __shared__ float sX[M_TILE * ROWPAD];    // 64 x 256 activation tile (padded)
    __shared__ float sPart[8][M_TILE];       // per-wave GEMV partials

    const int tid  = threadIdx.x;
    const int wave = tid >> 5;
    const int lane = tid & 31;
    const long long e0 = (long long)blockIdx.x * M_TILE;

    // ---- Gather + concat: stage X[64][256] into LDS with float4 traffic ----
    {
        const int m = tid >> 2;      // 0..63 : edge row in tile
        const int q = tid & 3;       // 0..3  : chunk phase
        long long e = e0 + m;
        int ru = 0, cbz = 0;
        if (e < E) { ru = eidx[e]; cbz = eidx[(long long)E + e]; }
        const float4* __restrict__ pu = (const float4*)(zu + (long long)ru  * HID);
        const float4* __restrict__ pb = (const float4*)(zb + (long long)cbz * HID);
        float* dst = sX + m * ROWPAD;
        #pragma unroll
        for (int i = 0; i < 16; ++i) {
            int c = q + i * 4;                      // float4 chunk 0..63 within row
            float4 v = (c < 32) ? pu[c] : pb[c - 32];
            *((float4*)(dst + c * 4)) = v;          // ds_store_b128 (16B aligned)
        }
    }
    __syncthreads();

    // ---- GEMM: H1 = relu(X @ W1 + b1). Wave w owns N-slice [16w,16w+16). ----
    // A 16x4 layout: lanes 0-15 hold M=lane; VGPR0 = K {0 | 2}, VGPR1 = K {1 | 3}.
    // B 4x16 layout (mirror): lane (n) striped; VGPR0 = K {0 | 2}, VGPR1 = K {1 | 3}.
    const int n0 = wave * 16;
    const int nl = n0 + (lane & 15);         // this lane's hidden column
    const int kb = (lane >> 4) * 2;          // K sub-offset 0 or 2

    v8f c0 = {}, c1 = {}, c2 = {}, c3 = {};  // edge groups [0,16) [16,32) [32,48) [48,64)

    const float* __restrict__ Bbase = W1 + (long long)kb * HID + nl;
    const float* A0 = sX + ((lane & 15) +  0) * ROWPAD + kb;
    const float* A1 = sX + ((lane & 15) + 16) * ROWPAD + kb;
    const float* A2 = sX + ((lane & 15) + 32) * ROWPAD + kb;
    const float* A3 = sX + ((lane & 15) + 48) * ROWPAD + kb;

    #pragma unroll 4
    for (int k0 = 0; k0 < TWOH; k0 += 4) {
        v2f bf;
        bf.x = Bbase[(long long)k0 * HID];           // W1[k0+kb  ][nl]
        bf.y = Bbase[(long long)k0 * HID + HID];     // W1[k0+kb+1][nl]
        v2f a0 = *(const v2f*)(A0 + k0);             // ds_load_b64 (8B aligned)
        v2f a1 = *(const v2f*)(A1 + k0);
        v2f a2 = *(const v2f*)(A2 + k0);
        v2f a3 = *(const v2f*)(A3 + k0);
        c0 = __builtin_amdgcn_wmma_f32_16x16x4_f32(false, a0, false, bf,
                                                   (short)0, c0, false, false);
        c1 = __builtin_amdgcn_wmma_f32_16x16x4_f32(false, a1, false, bf,
                                                   (short)0, c1, false, false);
        c2 = __builtin_amdgcn_wmma_f32_16x16x4_f32(false, a2, false, bf,
                                                   (short)0, c2, false, false);
        c3 = __builtin_amdgcn_wmma_f32_16x16x4_f32(false, a3, false, bf,
                                                   (short)0, c3, false, false);
    }

    // ---- Fused epilogue: relu(+b1), scale by W2[n], reduce over n per edge ----
    const float bv  = b1[nl];
    const float w2v = W2[nl];

    #pragma unroll
    for (int r = 0; r < 8; ++r) {
        float h0 = c0[r] + bv; h0 = h0 > 0.f ? h0 : 0.f;
        float h1 = c1[r] + bv; h1 = h1 > 0.f ? h1 : 0.f;
        float h2 = c2[r] + bv; h2 = h2 > 0.f ? h2 : 0.f;
        float h3 = c3[r] + bv; h3 = h3 > 0.f ? h3 : 0.f;
        float v0 = h0 * w2v, v1 = h1 * w2v, v2 = h2 * w2v, v3 = h3 * w2v;
        // Butterfly sum across the 16 lanes that share one edge row
        #pragma unroll
        for (int mask = 1; mask < 16; mask <<= 1) {
            v0 += __shfl_xor(v0, mask, 32);
            v1 += __shfl_xor(v1, mask, 32);
            v2 += __shfl_xor(v2, mask, 32);
            v3 += __shfl_xor(v3, mask, 32);
        }
        if ((lane & 15) == 0) {
            int m = r + 8 * (lane >> 4);              // C layout: M = r + 8*hi
            sPart[wave][ 0 + m] = v0;
            sPart[wave][16 + m] = v1;
            sPart[wave][32 + m] = v2;
            sPart[wave][48 + m] = v3;
        }
    }
    __syncthreads();

    // ---- Deterministic cross-wave sum (each wave covered a disjoint n-range) ----
    if (tid < M_TILE) {
        long long e = e0 + tid;
        if (e < E) {
            float s = b2[0];
            #pragma unroll
            for (int w = 0; w < 8; ++w) s += sPart[w][tid];
            out[e] = s;
        }
    }
}

extern "C" void kernel_launch(void* const* d_in, const int* in_sizes, int n_in,
                              void* d_out, int out_size, void* d_ws, size_t ws_size,
                              hipStream_t stream) {
    (void)n_in; (void)d_ws; (void)ws_size;
    const float* zu   = (const float*)d_in[0];
    const float* zb   = (const float*)d_in[1];
    const int*   eidx = (const int*)  d_in[2];   // [2,E]
    const float* W1   = (const float*)d_in[3];
    const float* b1   = (const float*)d_in[4];
    const float* W2   = (const float*)d_in[5];
    const float* b2   = (const float*)d_in[6];
    float* out = (float*)d_out;

    const int E = in_sizes[2] / 2;
    const int grid = (E + M_TILE - 1) / M_TILE;
    EdgeDecoder_wmma_f32<<<grid, 256, 0, stream>>>(zu, zb, eidx, W1, b1, W2, b2,
                                                   out, E);
    (void)out_size;
}